// GATEncoder_35579509080710
// MI455X (gfx1250) — compile-verified
//
#include <hip/hip_runtime.h>
#include <math.h>

typedef __attribute__((ext_vector_type(2))) float v2f;
typedef __attribute__((ext_vector_type(8))) float v8f;

#define LRELU_SLOPE 0.2f

// ---------------------------------------------------------------------------
// float atomic max via ordered-int encoding (works for +/- values, init -inf)
// ---------------------------------------------------------------------------
__device__ __forceinline__ void atomicMaxF(float* addr, float v) {
    if (v >= 0.0f) atomicMax((int*)addr, __float_as_int(v));
    else           atomicMin((unsigned int*)addr, __float_as_uint(v));
}

// ---------------------------------------------------------------------------
// fill
// ---------------------------------------------------------------------------
__global__ void fill_kernel(float* __restrict__ p, float v, long n) {
    long i = (long)blockIdx.x * blockDim.x + threadIdx.x;
    if (i < n) p[i] = v;
}

// ---------------------------------------------------------------------------
// D[M x COLS] = A[M x K] * B[K x COLS], f32 via V_WMMA_F32_16X16X4_F32.
// One wave per 16x16 output tile; wave w of the block covers cols [16w,16w+16).
// Fragment layouts per CDNA5 ISA 7.12.2 (32-bit A 16x4 / B 4x16 / C 16x16).
// No EXEC divergence before the WMMA (row index clamped, store guarded after).
// ---------------------------------------------------------------------------
template<int COLS>
__global__ void gemm_wmma_f32(const float* __restrict__ A, const float* __restrict__ B,
                              float* __restrict__ D, int M, int K) {
    const int lane    = threadIdx.x & 31;
    const int wave    = threadIdx.x >> 5;      // n-tile within block
    const int m0      = blockIdx.x * 16;
    const int n0      = wave * 16;
    const int halfsel = (lane < 16) ? 0 : 2;   // lanes 16-31 hold K+2, K+3
    int row = m0 + (lane & 15);
    if (row >= M) row = M - 1;                 // clamp, keep all lanes active
    const int col = n0 + (lane & 15);

    v8f acc = {};
    const float* ap = A + (size_t)row * K + halfsel;
    const float* bp = B + (size_t)halfsel * COLS + col;
#pragma unroll 4
    for (int k = 0; k < K; k += 4) {
        v2f a, b;
        a.x = ap[k];
        a.y = ap[k + 1];
        b.x = bp[(size_t)k * COLS];
        b.y = bp[(size_t)(k + 1) * COLS];
        acc = __builtin_amdgcn_wmma_f32_16x16x4_f32(false, a, false, b,
                                                    (short)0, acc, false, false);
    }
    // C/D layout: VGPR r -> row m0+r (lanes 0-15) / m0+8+r (lanes 16-31)
    const int rbase = m0 + ((lane < 16) ? 0 : 8);
#pragma unroll
    for (int r = 0; r < 8; ++r) {
        if (rbase + r < M) D[(size_t)(rbase + r) * COLS + col] = acc[r];
    }
}

// ---------------------------------------------------------------------------
// Per-node attention logits: asrc[n,h] = <h[n,h,:], att_s[h,:]>, same for adst.
// Layout [N][HEADS*HID]; t = n*HEADS+h indexes a contiguous HID block.
// ---------------------------------------------------------------------------
template<int HEADS, int HID>
__global__ void node_att(const float* __restrict__ h, const float* __restrict__ att_s,
                         const float* __restrict__ att_d, float* __restrict__ asrc,
                         float* __restrict__ adst, int N) {
    int t = blockIdx.x * blockDim.x + threadIdx.x;
    if (t >= N * HEADS) return;
    const int hd = t % HEADS;
    const float* hp = h + (size_t)t * HID;
    const float* as = att_s + hd * HID;
    const float* ad = att_d + hd * HID;
    float ss = 0.f, dd = 0.f;
#pragma unroll
    for (int c = 0; c < HID; ++c) { float x = hp[c]; ss += x * as[c]; dd += x * ad[c]; }
    asrc[t] = ss;
    adst[t] = dd;
}

// ---------------------------------------------------------------------------
// Edge pass 1: e = leakyrelu(asrc[src]+adst[dst]); segment max via atomics.
// Edges [0,E) from edge_index, [E,Etot) are self loops src=dst=e-E.
// ---------------------------------------------------------------------------
template<int HEADS>
__global__ void edge_score_max(const float* __restrict__ asrc, const float* __restrict__ adst,
                               const int* __restrict__ src, const int* __restrict__ dst,
                               float* __restrict__ esc, float* __restrict__ emax,
                               int E, int Etot) {
    int e = blockIdx.x * blockDim.x + threadIdx.x;
    if (e >= Etot) return;
    int s, d;
    if (e < E) { s = src[e]; d = dst[e]; } else { s = e - E; d = s; }
#pragma unroll
    for (int h2 = 0; h2 < HEADS; ++h2) {
        float v = asrc[(size_t)s * HEADS + h2] + adst[(size_t)d * HEADS + h2];
        v = (v > 0.f) ? v : LRELU_SLOPE * v;
        esc[(size_t)e * HEADS + h2] = v;
        atomicMaxF(&emax[(size_t)d * HEADS + h2], v);
    }
}

// ---------------------------------------------------------------------------
// Edge pass 2: ex = exp(e - emax[dst]); denom[dst] += ex (stored in place).
// ---------------------------------------------------------------------------
template<int HEADS>
__global__ void edge_exp_sum(const int* __restrict__ dst,
                             float* __restrict__ esc, const float* __restrict__ emax,
                             float* __restrict__ denom, int E, int Etot) {
    int e = blockIdx.x * blockDim.x + threadIdx.x;
    if (e >= Etot) return;
    const int d = (e < E) ? dst[e] : (e - E);
#pragma unroll
    for (int h2 = 0; h2 < HEADS; ++h2) {
        float ex = __expf(esc[(size_t)e * HEADS + h2] - emax[(size_t)d * HEADS + h2]);
        esc[(size_t)e * HEADS + h2] = ex;
        atomicAdd(&denom[(size_t)d * HEADS + h2], ex);
    }
}

// ---------------------------------------------------------------------------
// Edge pass 3: agg[dst] += ex * h[src]  (division by denom deferred).
// One wave per edge; each lane owns CPL contiguous channels.
// ---------------------------------------------------------------------------
template<int HEADS, int HID, int CPL>
__global__ void edge_aggregate(const float* __restrict__ h, const float* __restrict__ esc,
                               const int* __restrict__ src, const int* __restrict__ dst,
                               float* __restrict__ agg, int E, int Etot) {
    const int lane = threadIdx.x & 31;
    int e = blockIdx.x * (blockDim.x >> 5) + (threadIdx.x >> 5);
    if (e >= Etot) return;
    int s, d;
    if (e < E) { s = src[e]; d = dst[e]; } else { s = e - E; d = s; }
    const int CH   = HEADS * HID;
    const int c0   = lane * CPL;
    const int head = c0 / HID;
    const float w  = esc[(size_t)e * HEADS + head];
    if (CPL == 4) {
        const float4 hv = *(const float4*)(h + (size_t)s * CH + c0);
        float* ag = agg + (size_t)d * CH + c0;
        atomicAdd(ag + 0, w * hv.x);
        atomicAdd(ag + 1, w * hv.y);
        atomicAdd(ag + 2, w * hv.z);
        atomicAdd(ag + 3, w * hv.w);
    } else {
        atomicAdd(agg + (size_t)d * CH + c0, w * h[(size_t)s * CH + c0]);
    }
}

// ---------------------------------------------------------------------------
// Finalize: out = act(agg / (denom + 1e-16) + bias); act = ELU or softplus.
// ---------------------------------------------------------------------------
template<int HEADS, int HID, bool SOFTPLUS>
__global__ void finalize(const float* __restrict__ agg, const float* __restrict__ denom,
                         const float* __restrict__ bias, float* __restrict__ out, int N) {
    const int CH = HEADS * HID;
    long t = (long)blockIdx.x * blockDim.x + threadIdx.x;
    if (t >= (long)N * CH) return;
    const int n  = (int)(t / CH);
    const int c  = (int)(t % CH);
    const int hd = c / HID;
    float v = agg[t] / (denom[(size_t)n * HEADS + hd] + 1e-16f) + bias[c];
    if (SOFTPLUS) {
        out[t] = (v > 0.f) ? (v + log1pf(__expf(-v))) : log1pf(__expf(v));
    } else {
        out[t] = (v > 0.f) ? v : (__expf(v) - 1.f);
    }
}

// ---------------------------------------------------------------------------
// Host orchestration
// ---------------------------------------------------------------------------
extern "C" void kernel_launch(void* const* d_in, const int* in_sizes, int n_in,
                              void* d_out, int out_size, void* d_ws, size_t ws_size,
                              hipStream_t stream) {
    const float* x   = (const float*)d_in[0];
    const int*   ei  = (const int*)d_in[1];
    const float* W1  = (const float*)d_in[2];
    const float* as1 = (const float*)d_in[3];
    const float* ad1 = (const float*)d_in[4];
    const float* b1  = (const float*)d_in[5];
    const float* W2  = (const float*)d_in[6];
    const float* as2 = (const float*)d_in[7];
    const float* ad2 = (const float*)d_in[8];
    const float* b2  = (const float*)d_in[9];
    const float* W3  = (const float*)d_in[10];
    const float* as3 = (const float*)d_in[11];
    const float* ad3 = (const float*)d_in[12];
    const float* b3  = (const float*)d_in[13];

    const int N    = in_sizes[0] / 128;
    const int E    = in_sizes[1] / 2;
    const int Etot = E + N;
    const int* src = ei;
    const int* dst = ei + E;

    float* ws = (float*)d_ws;
    size_t o = 0;
    float* bufA = ws + o; o += (size_t)N * 128;
    float* bufB = ws + o; o += (size_t)N * 128;
    float* vAS  = ws + o; o += (size_t)N * 4;
    float* vAD  = ws + o; o += (size_t)N * 4;
    float* emax = ws + o; o += (size_t)N * 4;
    float* denm = ws + o; o += (size_t)N * 4;
    float* esc  = ws + o; o += (size_t)Etot * 4;
    float* h3   = ws + o; o += (size_t)N * 32;
    float* agg3 = ws + o; o += (size_t)N * 32;
    (void)ws_size; (void)n_in; (void)out_size;

    const int gM = (N + 15) / 16;                 // GEMM row tiles
    const int gE = (Etot + 255) / 256;            // thread-per-edge kernels
    const int gW = (Etot + 7) / 8;                // wave-per-edge (8 waves/block)
    const unsigned gN4   = (unsigned)(((long)N * 4   + 255) / 256);
    const unsigned gN128 = (unsigned)(((long)N * 128 + 255) / 256);
    const unsigned gN32  = (unsigned)(((long)N * 32  + 255) / 256);

    // ---- layer with HEADS=4, HID=32 ----
    auto layer4 = [&](const float* feat_in, const float* W, const float* as,
                      const float* ad, const float* bias,
                      float* hbuf, float* aggbuf, float* feat_out) {
        gemm_wmma_f32<128><<<gM, 256, 0, stream>>>(feat_in, W, hbuf, N, 128);
        node_att<4, 32><<<gN4, 256, 0, stream>>>(hbuf, as, ad, vAS, vAD, N);
        fill_kernel<<<gN4, 256, 0, stream>>>(emax, -INFINITY, (long)N * 4);
        fill_kernel<<<gN4, 256, 0, stream>>>(denm, 0.f, (long)N * 4);
        fill_kernel<<<gN128, 256, 0, stream>>>(aggbuf, 0.f, (long)N * 128);
        edge_score_max<4><<<gE, 256, 0, stream>>>(vAS, vAD, src, dst, esc, emax, E, Etot);
        edge_exp_sum<4><<<gE, 256, 0, stream>>>(dst, esc, emax, denm, E, Etot);
        edge_aggregate<4, 32, 4><<<gW, 256, 0, stream>>>(hbuf, esc, src, dst, aggbuf, E, Etot);
        finalize<4, 32, false><<<gN128, 256, 0, stream>>>(aggbuf, denm, bias, feat_out, N);
    };

    // Layer 1: x -> bufA(h) ; agg in bufB ; elu result back into bufA
    layer4(x, W1, as1, ad1, b1, bufA, bufB, bufA);
    // Layer 2: bufA -> bufB(h) ; agg in bufA ; elu result into bufB
    layer4(bufA, W2, as2, ad2, b2, bufB, bufA, bufB);

    // ---- layer 3: HEADS=1, HID=32 (=OUT), softplus output ----
    gemm_wmma_f32<32><<<gM, 64, 0, stream>>>(bufB, W3, h3, N, 128);
    node_att<1, 32><<<(unsigned)((N + 255) / 256), 256, 0, stream>>>(h3, as3, ad3, vAS, vAD, N);
    fill_kernel<<<(unsigned)((N + 255) / 256), 256, 0, stream>>>(emax, -INFINITY, N);
    fill_kernel<<<(unsigned)((N + 255) / 256), 256, 0, stream>>>(denm, 0.f, N);
    fill_kernel<<<gN32, 256, 0, stream>>>(agg3, 0.f, (long)N * 32);
    edge_score_max<1><<<gE, 256, 0, stream>>>(vAS, vAD, src, dst, esc, emax, E, Etot);
    edge_exp_sum<1><<<gE, 256, 0, stream>>>(dst, esc, emax, denm, E, Etot);
    edge_aggregate<1, 32, 1><<<gW, 256, 0, stream>>>(h3, esc, src, dst, agg3, E, Etot);
    finalize<1, 32, true><<<gN32, 256, 0, stream>>>(agg3, denm, b3, (float*)d_out, N);
}